// EquivariantPolicy_16295105921113
// MI455X (gfx1250) — compile-verified
//
#include <hip/hip_runtime.h>

// ---------------------------------------------------------------------------
// EquivariantPolicy for MI455X (gfx1250, wave32, WMMA).
//
// Math collapse: gmap[u,p] = p^-1 * u and featperm is a group homomorphism
// to feature-bit permutations, so
//   e[u] = (1/G) * X * Pi_u * Wsum + bias_mean,
//   Wsum[q,h] = sum_g eq_weight[g, featperm[g][q], h]
// and Pi_u * Wsum is Wsum with rows gathered by featperm[inv(u)].
// The mean over NACT commutes through Wout: m = h2 . wbar + bbar.
// Final: y[bt,c] = (1/6) * sum_{u in coset c} m[u,bt].
//
// B matrices are stored K-contiguous ("transposed") in LDS so every WMMA
// fragment (A and B) is one aligned ds_load_b64 -> no register glue movs.
// ---------------------------------------------------------------------------

constexpr int kQ    = 64;
constexpr int kG    = 120;
constexpr int kHID  = 128;
constexpr int kNACT = 20;
constexpr int kBT   = 2048;          // B*T = 32*64
constexpr int kMT   = 128;           // rows per block tile -> 16 tiles

// LDS strides (floats).  All even (8B-aligned b64 fragments) and == 4 mod 64
// so the 16 lanes of each wave-half hit disjoint mod-4 bank classes.
constexpr int XS_STRIDE  = 68;       // Xs  [row][k]   (A, GEMM1)
constexpr int WST_STRIDE = 68;       // WsT [col][k]   (B, GEMM1, transposed)
constexpr int HS_STRIDE  = 132;      // Hs  [row][k]   (A, GEMM2)
constexpr int W1T_STRIDE = 132;      // W1T [col][k]   (B, GEMM2, transposed)

typedef float v2f __attribute__((ext_vector_type(2)));
typedef float v8f __attribute__((ext_vector_type(8)));

// ---------------------------------------------------------------------------
// Compile-time group tables.
// ---------------------------------------------------------------------------
struct Tables {
  int featperm[kG * kQ];   // featperm[p][n]
  int invp[kG];            // rank of p^-1
  int cmem[kNACT * 6];     // 6 members of each coset
};

constexpr Tables make_tables() {
  Tables T{};
  int perms[kG][5] = {};
  const int fact[5] = {24, 6, 2, 1, 1};
  for (int t = 0; t < kG; ++t) {
    int avail[5] = {0, 1, 2, 3, 4};
    int rem = t;
    for (int k = 0; k < 5; ++k) {
      int idx = rem / fact[k]; rem -= idx * fact[k];
      perms[t][k] = avail[idx];
      for (int j = idx; j < 4 - k; ++j) avail[j] = avail[j + 1];
    }
    int ip[5] = {};
    for (int k = 0; k < 5; ++k) ip[perms[t][k]] = k;
    int rank = 0;
    for (int k = 0; k < 5; ++k) {
      int c = 0;
      for (int j = k + 1; j < 5; ++j) c += (ip[j] < ip[k]) ? 1 : 0;
      rank += c * fact[k];
    }
    T.invp[t] = rank;
    for (int n = 0; n < 64; ++n) {
      int v = n & 32;
      for (int k = 0; k < 5; ++k)
        v |= ((n >> (4 - k)) & 1) << (4 - perms[t][k]);
      T.featperm[t * 64 + n] = v;
    }
  }
  int cnt[kNACT] = {};
  for (int u = 0; u < kG; ++u) {
    const int p0 = perms[u][0], p1 = perms[u][1];
    const int c = p0 * 4 + (p1 > p0 ? p1 - 1 : p1);
    T.cmem[c * 6 + cnt[c]] = u;
    cnt[c]++;
  }
  return T;
}

__device__ constexpr Tables g_tables = make_tables();

// ---- workspace layout (bytes) ----
constexpr size_t OFF_WSUM     = 0;        // float[64*128]
constexpr size_t OFF_BIASMEAN = 32768;    // float[128]
constexpr size_t OFF_WBARB    = 33280;    // float[129]  (wbar[128], bbar)
constexpr size_t OFF_M        = 34048;    // float[120*2048]

// ---------------------------------------------------------------------------
// Kernel B: fold the weights.  Wsum, bias mean, wbar = Wout.mean(1), bbar.
// ---------------------------------------------------------------------------
__global__ void build_consts_kernel(const float* __restrict__ eqw,
                                    const float* __restrict__ eqb,
                                    const float* __restrict__ Wout,
                                    const float* __restrict__ bout,
                                    float* __restrict__ Wsum,
                                    float* __restrict__ biasmean,
                                    float* __restrict__ wbarb) {
  const int bid = blockIdx.x;
  const int tid = threadIdx.x;
  if (bid < 32) {
    const int idx = bid * 256 + tid;               // 0..8191
    const int q = idx >> 7, h = idx & 127;
    float s = 0.f;
    for (int g = 0; g < kG; ++g)
      s += eqw[(size_t)(g * kQ + g_tables.featperm[g * kQ + q]) * kHID + h];
    Wsum[q * kHID + h] = s;
  } else {
    if (tid < 128) {
      float s = 0.f;
      for (int g = 0; g < kG; ++g) s += eqb[g * kHID + tid];
      biasmean[tid] = s * (1.0f / kG);
    } else {
      const int h = tid - 128;
      float s = 0.f;
      for (int a = 0; a < kNACT; ++a) s += Wout[h * kNACT + a];
      wbarb[h] = s * (1.0f / kNACT);
      if (h == 0) {
        float sb = 0.f;
        for (int a = 0; a < kNACT; ++a) sb += bout[a];
        wbarb[128] = sb * (1.0f / kNACT);
      }
    }
  }
}

// ---------------------------------------------------------------------------
// Kernel C: per (u, 128-row tile):  e = relu(X @ Ws_u / G + bmean),
//           h2 = relu(e @ W1 + b1),  m = h2 . wbar + bbar   (fused).
// 256 threads = 8 wave32; each wave owns a 16-row stripe, N=128 as 8 WMMA
// column tiles, fp32 V_WMMA_F32_16X16X4_F32 accumulation.
// ---------------------------------------------------------------------------
__global__ void __launch_bounds__(256)
eqpolicy_main_kernel(const float* __restrict__ x,
                     const float* __restrict__ W1,
                     const float* __restrict__ b1,
                     const float* __restrict__ Wsum,
                     const float* __restrict__ biasmean,
                     const float* __restrict__ wbarb,
                     float* __restrict__ mws) {
  extern __shared__ float smem[];
  float* Xs  = smem;                               // [128][XS_STRIDE]
  float* WsT = Xs  + kMT  * XS_STRIDE;             // [128 cols][WST_STRIDE]
  float* Hs  = WsT + kHID * WST_STRIDE;            // [128][HS_STRIDE]
  float* W1T = Hs  + kMT  * HS_STRIDE;             // [128 cols][W1T_STRIDE]

  const int tid   = threadIdx.x;
  const int lane  = tid & 31;
  const int w     = tid >> 5;                      // wave id 0..7
  const int m     = lane & 15;                     // M (A) / N (B,C) index
  const int half  = lane >> 4;                     // 0 | 1
  const int khalf = half * 2;                      // A/B K sub-offset
  const int u     = blockIdx.y;
  const int row0  = blockIdx.x * kMT;

  // ---- stage X tile (straight, coalesced float4) ----
  {
    const float* xb = x + (size_t)row0 * kQ;
    for (int i = tid; i < kMT * (kQ / 4); i += 256) {
      const int r = i >> 4, c4 = i & 15;
      const float4 v = ((const float4*)(xb + r * kQ))[c4];
      ((float4*)(Xs + r * XS_STRIDE))[c4] = v;
    }
  }
  // ---- stage WsT[col][k] = Wsum[featperm[inv(u)][k]][col]  (transposed) ----
  {
    const int invu = g_tables.invp[u];
    const int* fpi = g_tables.featperm + invu * kQ;
    for (int i = tid; i < kQ * kHID; i += 256) {   // coalesced global reads
      const int r = i >> 7, c = i & 127;           // r = k row, c = column
      WsT[c * WST_STRIDE + r] = Wsum[(size_t)fpi[r] * kHID + c];
    }
  }
  // ---- stage W1T[col][k] = W1[k][col]  (transposed) ----
  {
    for (int i = tid; i < kHID * kHID; i += 256) { // coalesced global reads
      const int r = i >> 7, c = i & 127;
      W1T[c * W1T_STRIDE + r] = W1[r * kHID + c];
    }
  }
  __syncthreads();

  const v8f zero = {0.f, 0.f, 0.f, 0.f, 0.f, 0.f, 0.f, 0.f};
  v8f acc[8];

  // ================= GEMM1: e = Xs(128x64) @ Ws(64x128) =================
#pragma unroll
  for (int n = 0; n < 8; ++n) acc[n] = zero;
  {
    const float* Arow = Xs + (w * 16 + m) * XS_STRIDE + khalf;
    const float* Bcol = WsT + m * WST_STRIDE + khalf;
    for (int k0 = 0; k0 < kQ; k0 += 4) {
      const v2f a = *(const v2f*)(Arow + k0);      // one ds_load_b64
#pragma unroll
      for (int n = 0; n < 8; ++n) {
        const v2f b = *(const v2f*)(Bcol + n * 16 * WST_STRIDE + k0);
        acc[n] = __builtin_amdgcn_wmma_f32_16x16x4_f32(
            false, a, false, b, (short)0, acc[n], false, false);
      }
    }
  }
  // epilogue 1: scale, bias, relu -> Hs (wave writes only its own 16 rows)
#pragma unroll
  for (int n = 0; n < 8; ++n) {
    const int col = n * 16 + m;
    const float bm = biasmean[col];
#pragma unroll
    for (int r = 0; r < 8; ++r) {
      const int row = w * 16 + half * 8 + r;
      const float v = acc[n][r] * (1.0f / kG) + bm;
      Hs[row * HS_STRIDE + col] = fmaxf(v, 0.0f);
    }
  }
  // no __syncthreads needed: each wave consumes exactly the rows it produced

  // ================= GEMM2: h2 = Hs(128x128) @ W1(128x128) =================
#pragma unroll
  for (int n = 0; n < 8; ++n) acc[n] = zero;
  {
    const float* Arow = Hs + (w * 16 + m) * HS_STRIDE + khalf;
    const float* Bcol = W1T + m * W1T_STRIDE + khalf;
    for (int k0 = 0; k0 < kHID; k0 += 4) {
      const v2f a = *(const v2f*)(Arow + k0);
#pragma unroll
      for (int n = 0; n < 8; ++n) {
        const v2f b = *(const v2f*)(Bcol + n * 16 * W1T_STRIDE + k0);
        acc[n] = __builtin_amdgcn_wmma_f32_16x16x4_f32(
            false, a, false, b, (short)0, acc[n], false, false);
      }
    }
  }

  // epilogue 2 (fused): m[row] = sum_col relu(acc + b1[col]) * wbar[col] + bbar
  float rowacc[8] = {0.f, 0.f, 0.f, 0.f, 0.f, 0.f, 0.f, 0.f};
#pragma unroll
  for (int n = 0; n < 8; ++n) {
    const int col = n * 16 + m;
    const float wb = wbarb[col];
    const float bb = b1[col];
#pragma unroll
    for (int r = 0; r < 8; ++r)
      rowacc[r] += fmaxf(acc[n][r] + bb, 0.0f) * wb;
  }
  // butterfly-reduce across the 16 lanes of each half (wave32)
#pragma unroll
  for (int s = 8; s >= 1; s >>= 1) {
#pragma unroll
    for (int r = 0; r < 8; ++r)
      rowacc[r] += __shfl_xor(rowacc[r], s, 32);
  }
  if (m == 0) {
    const float bbar = wbarb[128];
    const int rbase = row0 + w * 16 + half * 8;
#pragma unroll
    for (int r = 0; r < 8; ++r)
      mws[(size_t)u * kBT + rbase + r] = rowacc[r] + bbar;
  }
}

// ---------------------------------------------------------------------------
// Kernel D: coset segment-sum (every coset has exactly 3! = 6 members).
// ---------------------------------------------------------------------------
__global__ void reduce_out_kernel(const float* __restrict__ mws,
                                  float* __restrict__ out) {
  const int idx = blockIdx.x * blockDim.x + threadIdx.x;
  if (idx >= kBT * kNACT) return;
  const int bt = idx / kNACT, c = idx % kNACT;
  float s = 0.f;
#pragma unroll
  for (int j = 0; j < 6; ++j)
    s += mws[(size_t)g_tables.cmem[c * 6 + j] * kBT + bt];
  out[bt * kNACT + c] = s * (1.0f / 6.0f);
}

// ---------------------------------------------------------------------------
extern "C" void kernel_launch(void* const* d_in, const int* in_sizes, int n_in,
                              void* d_out, int out_size, void* d_ws, size_t ws_size,
                              hipStream_t stream) {
  const float* x    = (const float*)d_in[0];
  const float* eqw  = (const float*)d_in[1];
  const float* eqb  = (const float*)d_in[2];
  const float* W1   = (const float*)d_in[3];
  const float* b1   = (const float*)d_in[4];
  const float* Wout = (const float*)d_in[5];
  const float* bout = (const float*)d_in[6];
  float* out = (float*)d_out;

  char* ws = (char*)d_ws;
  float* Wsum     = (float*)(ws + OFF_WSUM);
  float* biasmean = (float*)(ws + OFF_BIASMEAN);
  float* wbarb    = (float*)(ws + OFF_WBARB);
  float* mws      = (float*)(ws + OFF_M);

  build_consts_kernel<<<33, 256, 0, stream>>>(eqw, eqb, Wout, bout,
                                              Wsum, biasmean, wbarb);

  const size_t smem_bytes =
      (size_t)(kMT * XS_STRIDE + kHID * WST_STRIDE +
               kMT * HS_STRIDE + kHID * W1T_STRIDE) * sizeof(float);
  dim3 grid(kBT / kMT, kG);   // 16 x 120
  eqpolicy_main_kernel<<<grid, 256, smem_bytes, stream>>>(
      x, W1, b1, Wsum, biasmean, wbarb, mws);

  reduce_out_kernel<<<(kBT * kNACT + 255) / 256, 256, 0, stream>>>(mws, out);
}